// SE_34720515621164
// MI455X (gfx1250) — compile-verified
//
#include <hip/hip_runtime.h>
#include <hip/hip_bf16.h>
#include <math.h>

// SE block: x[64,128,112,112] fp32
//   y = mean(x, HW); y = mish(y@W1^T + b1); y = sigmoid(y@W2^T + b2); out = x*y
//
// Kernel 1: per-(b,c) plane mean (HBM-bound, NT loads: 411MB read)
// Kernel 2: tiny MLP via V_WMMA_F32_16X16X4_F32 (fp32 WMMA, exact vs reference)
//           W1 zero-padded into LDS -> branch-free WMMA inner loops
// Kernel 3: broadcast scale (HBM-bound, NT load+store: 411MB + 411MB)

#define B_   64
#define C_   128
#define HID_ 8
#define HW_  12544            // 112*112
#define HW4_ (HW_ / 4)        // 3136 float4 per plane
#define NTOT4_ ((size_t)B_ * C_ * HW4_)   // 25,690,112 float4

typedef __attribute__((ext_vector_type(2))) float v2f;
typedef __attribute__((ext_vector_type(4))) float v4f;   // native vec for NT ld/st
typedef __attribute__((ext_vector_type(8))) float v8f;

// ---------------------------------------------------------------- kernel 1
__global__ void se_squeeze_kernel(const float* __restrict__ x,
                                  float* __restrict__ y) {
    // one block per (b,c) plane; blockIdx.x == b*C_ + c
    const v4f* __restrict__ p =
        (const v4f*)(x + (size_t)blockIdx.x * HW_);

    float sum = 0.0f;
    for (int i = threadIdx.x; i < HW4_; i += 256) {
        v4f v = __builtin_nontemporal_load(&p[i]);   // streaming: no reuse
        sum += (v.x + v.y) + (v.z + v.w);
    }

    // wave32 reduction
    #pragma unroll
    for (int off = 16; off > 0; off >>= 1)
        sum += __shfl_xor(sum, off, 32);

    __shared__ float wsum[8];
    const int lane = threadIdx.x & 31;
    const int wid  = threadIdx.x >> 5;
    if (lane == 0) wsum[wid] = sum;
    __syncthreads();

    if (threadIdx.x == 0) {
        float t = 0.0f;
        #pragma unroll
        for (int i = 0; i < 8; ++i) t += wsum[i];
        y[blockIdx.x] = t * (1.0f / (float)HW_);
    }
}

// ---------------------------------------------------------------- kernel 2
__device__ __forceinline__ float mishf(float v) {
    // v * tanh(softplus(v)); guard exp overflow (softplus(v) ~= v for large v)
    float sp = (v > 20.0f) ? v : log1pf(expf(v));
    return v * tanhf(sp);
}

__global__ void se_mlp_wmma_kernel(const float* __restrict__ y,   // [64,128]
                                   const float* __restrict__ W1,  // [8,128]
                                   const float* __restrict__ b1,  // [8]
                                   const float* __restrict__ W2,  // [128,8]
                                   const float* __restrict__ b2,  // [128]
                                   float* __restrict__ s) {       // [64,128]
    // 128 threads = 4 wave32 waves; wave w owns output rows [16w, 16w+16)
    const int lane = threadIdx.x & 31;
    const int wid  = threadIdx.x >> 5;
    const int m0   = wid * 16;
    const bool hi  = lane >= 16;
    const int l15  = lane & 15;
    const int row  = m0 + l15;          // A-matrix row for this lane
    const int khat = hi ? 2 : 0;        // A/B frags: hi half-wave holds K+2,K+3

    __shared__ float w1p[16 * C_];      // W1^T source, zero-padded rows 8..15
    __shared__ float act[B_ * HID_];    // mish activations [64][8]

    // stage W1 zero-padded into LDS (uniform, branch-free WMMA loops after)
    for (int i = threadIdx.x; i < 16 * C_; i += 128) {
        const int n = i >> 7;           // padded output channel 0..15
        const int k = i & (C_ - 1);
        w1p[i] = (n < HID_) ? W1[n * C_ + k] : 0.0f;
    }
    __syncthreads();

    // ---- layer 1: D[64,16] = y[64,128] @ W1^T(padded 128x16), K=128 ----
    {
        v8f acc = {};
        #pragma unroll 4
        for (int k0 = 0; k0 < C_; k0 += 4) {
            const int ka = k0 + khat;
            v2f a, b;
            a.x = y[row * C_ + ka];
            a.y = y[row * C_ + ka + 1];
            // B[k][n] = W1pad[n][k]
            b.x = w1p[l15 * C_ + ka];
            b.y = w1p[l15 * C_ + ka + 1];
            acc = __builtin_amdgcn_wmma_f32_16x16x4_f32(
                false, a, false, b, (short)0, acc, false, false);
        }
        // D layout: vgpr r -> row m0+r (lanes 0-15) / m0+8+r (lanes 16-31),
        // col = lane&15
        if (l15 < HID_) {
            #pragma unroll
            for (int r = 0; r < 8; ++r) {
                const int rd = m0 + r + (hi ? 8 : 0);
                float v = acc[r] + b1[l15];
                act[rd * HID_ + l15] = mishf(v);
            }
        }
    }
    __syncthreads();

    // ---- layer 2: D[64,128] = act[64,8] @ W2^T[8,128], K=8 (2 steps) ----
    for (int nt = 0; nt < C_ / 16; ++nt) {
        const int n0 = nt * 16;
        const int n  = n0 + l15;
        v8f acc = {};
        #pragma unroll
        for (int k0 = 0; k0 < HID_; k0 += 4) {
            const int ka = k0 + khat;
            v2f a, b;
            a.x = act[row * HID_ + ka];
            a.y = act[row * HID_ + ka + 1];
            // B[k][n] = W2[n][k]
            b.x = W2[n * HID_ + ka];
            b.y = W2[n * HID_ + ka + 1];
            acc = __builtin_amdgcn_wmma_f32_16x16x4_f32(
                false, a, false, b, (short)0, acc, false, false);
        }
        #pragma unroll
        for (int r = 0; r < 8; ++r) {
            const int rd = m0 + r + (hi ? 8 : 0);
            float v = acc[r] + b2[n];
            s[rd * C_ + n] = 1.0f / (1.0f + expf(-v));
        }
    }
}

// ---------------------------------------------------------------- kernel 3
__global__ void se_scale_kernel(const float* __restrict__ x,
                                const float* __restrict__ s,
                                float* __restrict__ out) {
    const size_t idx = (size_t)blockIdx.x * 256 + threadIdx.x;   // float4 id
    if (idx >= NTOT4_) return;
    // 12544 % 4 == 0 -> all 4 elements share one (b,c); plane = 3136 float4
    const int bc = (int)(idx / (size_t)HW4_);
    const float sc = s[bc];                 // small table: stays cache-resident
    v4f v = __builtin_nontemporal_load(&((const v4f*)x)[idx]);
    v.x *= sc; v.y *= sc; v.z *= sc; v.w *= sc;
    __builtin_nontemporal_store(v, &((v4f*)out)[idx]);
}

// ---------------------------------------------------------------- launch
extern "C" void kernel_launch(void* const* d_in, const int* in_sizes, int n_in,
                              void* d_out, int out_size, void* d_ws, size_t ws_size,
                              hipStream_t stream) {
    const float* x  = (const float*)d_in[0];
    const float* W1 = (const float*)d_in[1];
    const float* b1 = (const float*)d_in[2];
    const float* W2 = (const float*)d_in[3];
    const float* b2 = (const float*)d_in[4];
    float* out = (float*)d_out;

    float* y = (float*)d_ws;        // [64,128] pooled means
    float* s = y + B_ * C_;         // [64,128] sigmoid scales

    se_squeeze_kernel<<<B_ * C_, 256, 0, stream>>>(x, y);
    se_mlp_wmma_kernel<<<1, 128, 0, stream>>>(y, W1, b1, W2, b2, s);

    const int nblk = (int)((NTOT4_ + 255) / 256);   // 100,352
    se_scale_kernel<<<nblk, 256, 0, stream>>>(x, s, out);
}